// Attention_21706764714054
// MI455X (gfx1250) — compile-verified
//
#include <hip/hip_runtime.h>
#include <math.h>

// ---------------------------------------------------------------------------
// Coverage-attention forward for MI455X (gfx1250), fp32 end-to-end.
//  - All matmuls on V_WMMA_F32_16X16X4_F32 (exact fp32 matrix pipe).
//  - Topic slabs staged with GLOBAL_LOAD_ASYNC_TO_LDS_B128 (ASYNCcnt),
//    double-buffered against the q slab => async copy fully overlaps the
//    4096-WMMA q@Wa^T phase. s_wait_asynccnt 0 gates the pk GEMM.
//  - Wave-private LDS slabs => no workgroup barriers anywhere.
// ---------------------------------------------------------------------------

typedef float v2f __attribute__((ext_vector_type(2)));
typedef float v8f __attribute__((ext_vector_type(8)));

#define B_N    16384
#define K_T    5
#define H_D    512
#define E_D    512
#define PITCH  516                 // LDS row pitch in floats (bank-conflict pad)
#define RPW    16                  // batch rows per wave
#define WPG    2                   // waves per workgroup
#define RPG    (RPW * WPG)         // 32 rows per workgroup
#define NCHUNK (H_D / 16)          // 32 output chunks of 16 columns

#if defined(__gfx1250__)
// Async 16B/lane copy global -> LDS, tracked by ASYNCcnt.
__device__ __forceinline__ void async_ld_b128(const float* g, float* l) {
  unsigned lofs = (unsigned)(uintptr_t)l;              // flat->LDS: low 32 bits
  unsigned long long ga = (unsigned long long)(uintptr_t)g;
  asm volatile("global_load_async_to_lds_b128 %0, %1, off"
               :: "v"(lofs), "v"(ga) : "memory");
}
__device__ __forceinline__ void wait_async0() {
  asm volatile("s_wait_asynccnt 0x0" ::: "memory");
}
#endif

__global__ __launch_bounds__(WPG * 32, 1)
void attn_cov_kernel(const float* __restrict__ query,
                     const float* __restrict__ topics,
                     const float* __restrict__ cov,
                     const float* __restrict__ Ua,
                     const float* __restrict__ Wa,
                     const float* __restrict__ va_w,
                     const float* __restrict__ va_b,
                     float* __restrict__ out_mt,
                     float* __restrict__ out_al)
{
#if defined(__gfx1250__)
#if !__has_builtin(__builtin_amdgcn_wmma_f32_16x16x4_f32)
#error "gfx1250 device pass: __builtin_amdgcn_wmma_f32_16x16x4_f32 not available"
#else
  __shared__ float lds[WPG * 2 * RPW * PITCH];  // per wave: q slab + t slab (2x33KB)
  __shared__ float s_lds[RPG][8];               // raw scores, padded row

  const int lane = threadIdx.x & 31;
  const int wave = threadIdx.x >> 5;
  const int n15  = lane & 15;          // M (A-frag) / N (B,C-frag) selector
  const int hi   = lane >> 4;          // half-wave id (K-pair / M+8 selector)
  const int row0 = blockIdx.x * RPG + wave * RPW;

  float* qbuf = lds + wave * 2 * RPW * PITCH;   // 16 x 512 q slab (A-operand)
  float* tbuf = qbuf + RPW * PITCH;             // 16 x 512 topic slab

  // ---- stage query rows (initial q) into q slab (sync: DS-ordered w/ A) --
  for (int idx = lane; idx < RPW * (H_D / 4); idx += 32) {
    const int r  = idx >> 7;           // / 128
    const int c4 = idx & 127;
    const float4 v = *(const float4*)(query + (size_t)(row0 + r) * H_D + c4 * 4);
    *(float4*)(qbuf + r * PITCH + c4 * 4) = v;
  }

  // ---- kick off async staging of topics[:, 0, :] into t slab -------------
  {
    const float* t0 = topics + (size_t)row0 * (K_T * E_D);
    for (int idx = lane; idx < RPW * (E_D / 4); idx += 32) {
      const int r  = idx >> 7;
      const int c4 = idx & 127;
      async_ld_b128(t0 + (size_t)r * (K_T * E_D) + c4 * 4,
                    tbuf + r * PITCH + c4 * 4);
    }
  }

  v8f accQ[NCHUNK];
  const v8f vzero = {0.f, 0.f, 0.f, 0.f, 0.f, 0.f, 0.f, 0.f};

  for (int i = 0; i < K_T; ++i) {
    #pragma unroll
    for (int c = 0; c < NCHUNK; ++c) accQ[c] = vzero;

    // ---- Phase A: q_next = q @ Wa^T  (A from LDS, Wa columns from L2) ----
    for (int k0 = 0; k0 < H_D; k0 += 4) {
      const v2f a = *(const v2f*)(qbuf + n15 * PITCH + k0 + 2 * hi);
      #pragma unroll
      for (int c = 0; c < NCHUNK; ++c) {
        const v2f b = *(const v2f*)(Wa + (size_t)(c * 16 + n15) * H_D + k0 + 2 * hi);
        accQ[c] = __builtin_amdgcn_wmma_f32_16x16x4_f32(
            false, a, false, b, (short)0, accQ[c], false, false);
      }
    }

    // ---- topic slab must be resident before the pk GEMM ------------------
    wait_async0();

    // ---- Phase C: pk = t_i @ Ua^T, fused tanh + dot(va_w) ----------------
    float sAcc[8];
    #pragma unroll
    for (int r = 0; r < 8; ++r) sAcc[r] = 0.f;

    #pragma unroll
    for (int c = 0; c < NCHUNK; ++c) {
      v8f accP = vzero;
      for (int k0 = 0; k0 < E_D; k0 += 4) {
        const v2f a = *(const v2f*)(tbuf + n15 * PITCH + k0 + 2 * hi);
        const v2f b = *(const v2f*)(Ua + (size_t)(c * 16 + n15) * E_D + k0 + 2 * hi);
        accP = __builtin_amdgcn_wmma_f32_16x16x4_f32(
            false, a, false, b, (short)0, accP, false, false);
      }
      const float vw = va_w[c * 16 + n15];
      #pragma unroll
      for (int r = 0; r < 8; ++r)
        sAcc[r] += vw * tanhf(accQ[c][r] + accP[r]);
    }

    // ---- reduce dot over the 16 lanes in each half; stash raw score ------
    #pragma unroll
    for (int r = 0; r < 8; ++r) {
      float v = sAcc[r];
      v += __shfl_xor(v, 1);
      v += __shfl_xor(v, 2);
      v += __shfl_xor(v, 4);
      v += __shfl_xor(v, 8);
      if (n15 == 0) s_lds[wave * RPW + hi * 8 + r][i] = v;
    }

    // ---- t slab consumed: launch async staging for step i+1 --------------
    if (i + 1 < K_T) {
      const float* tn = topics + ((size_t)row0 * K_T + (i + 1)) * E_D;
      for (int idx = lane; idx < RPW * (E_D / 4); idx += 32) {
        const int r  = idx >> 7;
        const int c4 = idx & 127;
        async_ld_b128(tn + (size_t)r * (K_T * E_D) + c4 * 4,
                      tbuf + r * PITCH + c4 * 4);
      }
    }

    // ---- Phase D: write q_next back to q slab (A-layout for next step) ---
    #pragma unroll
    for (int c = 0; c < NCHUNK; ++c) {
      #pragma unroll
      for (int r = 0; r < 8; ++r)
        qbuf[(size_t)(hi * 8 + r) * PITCH + c * 16 + n15] = accQ[c][r];
    }
  }

  // ---- epilogue: coverage * (score + b), softmax(K=5), alphas, mt --------
  const float vab = va_b[0];
  for (int r = 0; r < RPW; ++r) {
    const int b = row0 + r;
    float sc[K_T];
    float mx = -3.402823466e38f;
    #pragma unroll
    for (int k = 0; k < K_T; ++k) {
      sc[k] = (s_lds[wave * RPW + r][k] + vab) * cov[(size_t)b * K_T + k];
      mx = fmaxf(mx, sc[k]);
    }
    float sum = 0.f;
    #pragma unroll
    for (int k = 0; k < K_T; ++k) { sc[k] = __expf(sc[k] - mx); sum += sc[k]; }
    const float inv = 1.f / sum;
    if (lane < K_T) out_al[(size_t)b * K_T + lane] = sc[lane] * inv;

    for (int e = lane; e < E_D; e += 32) {
      float acc = 0.f;
      #pragma unroll
      for (int k = 0; k < K_T; ++k)
        acc += (sc[k] * inv) * topics[((size_t)b * K_T + k) * E_D + e];
      out_mt[(size_t)b * E_D + e] = acc;
    }
  }
#endif  // __has_builtin
#else
  // Host pass (and non-gfx1250 device passes): empty body so the host
  // compilation never sees device-target builtins / gfx1250 asm.
  (void)query; (void)topics; (void)cov; (void)Ua; (void)Wa;
  (void)va_w; (void)va_b; (void)out_mt; (void)out_al;
#endif
}

extern "C" void kernel_launch(void* const* d_in, const int* in_sizes, int n_in,
                              void* d_out, int out_size, void* d_ws, size_t ws_size,
                              hipStream_t stream) {
  (void)in_sizes; (void)n_in; (void)out_size; (void)d_ws; (void)ws_size;
  const float* query  = (const float*)d_in[0];
  const float* topics = (const float*)d_in[1];
  const float* cov    = (const float*)d_in[2];
  const float* Ua     = (const float*)d_in[3];
  const float* Wa     = (const float*)d_in[4];
  const float* va_w   = (const float*)d_in[5];
  const float* va_b   = (const float*)d_in[6];

  float* out_mt = (float*)d_out;                       // [B, E]
  float* out_al = out_mt + (size_t)B_N * E_D;          // [B, K]

  dim3 grid(B_N / RPG);   // 512 workgroups
  dim3 block(WPG * 32);   // 64 threads = 2 waves
  hipLaunchKernelGGL(attn_cov_kernel, grid, block, 0, stream,
                     query, topics, cov, Ua, Wa, va_w, va_b, out_mt, out_al);
}